// TAGGCN_55009941128033
// MI455X (gfx1250) — compile-verified
//
#include <hip/hip_runtime.h>

// TAGConv (2x TAGConv(k=3) + FC) for MI455X / gfx1250.
// Propagation = L2-resident scatter-add (atomic f32); GEMMs use
// V_WMMA_F32_16X16X4_F32 (full fp32, matches reference precision).

typedef __attribute__((ext_vector_type(2))) float v2f;
typedef __attribute__((ext_vector_type(8))) float v8f;

#define TAG_NODES 50000
#define TAG_EDGES 800000
#define TAG_D     128
#define FSTRIDE   512   // 4 hop slots of 128 features, row-major concat

// ---------------- elementwise helpers ----------------

__global__ void tag_zero_f32(float* __restrict__ p, int n) {
    int i = blockIdx.x * blockDim.x + threadIdx.x;
    if (i < n) p[i] = 0.0f;
}

__global__ void tag_deg(const int* __restrict__ dst, float* __restrict__ deg, int nE) {
    int i = blockIdx.x * blockDim.x + threadIdx.x;
    if (i < nE) atomicAdd(&deg[dst[i]], 1.0f);
}

__global__ void tag_norm_finalize(float* __restrict__ p, int n) {
    int i = blockIdx.x * blockDim.x + threadIdx.x;
    if (i < n) p[i] = rsqrtf(fmaxf(p[i], 1.0f));
}

// in: (nodes,128) contiguous -> slot (stride FSTRIDE)
__global__ void tag_copy_to_slot(const float* __restrict__ in, float* __restrict__ slot, int nElems) {
    int i = blockIdx.x * blockDim.x + threadIdx.x;
    if (i < nElems) {
        int r = i >> 7, c = i & 127;
        slot[(size_t)r * FSTRIDE + c] = in[(size_t)r * TAG_D + c];
    }
}

__global__ void tag_zero_slot(float* __restrict__ slot, int nElems) {
    int i = blockIdx.x * blockDim.x + threadIdx.x;
    if (i < nElems) {
        int r = i >> 7, c = i & 127;
        slot[(size_t)r * FSTRIDE + c] = 0.0f;
    }
}

__global__ void tag_scale_slot(float* __restrict__ slot, const float* __restrict__ norm, int nElems) {
    int i = blockIdx.x * blockDim.x + threadIdx.x;
    if (i < nElems) {
        int r = i >> 7, c = i & 127;
        slot[(size_t)r * FSTRIDE + c] *= norm[r];
    }
}

// ---------------- edge scatter: one wave (32 lanes) per edge ----------------
// cur[dst] += prev[src] * norm[src]   (128 features, float4 per lane)

__global__ void tag_edge_scatter(const int* __restrict__ src, const int* __restrict__ dst,
                                 const float* __restrict__ norm,
                                 const float* __restrict__ prev, float* __restrict__ cur,
                                 int nE) {
    int gid  = blockIdx.x * blockDim.x + threadIdx.x;
    int e    = gid >> 5;
    int lane = gid & 31;
    if (e >= nE) return;
    int s = src[e];
    int d = dst[e];
    float ns = norm[s];
    const float4* hp = (const float4*)(prev + (size_t)s * FSTRIDE);
    float4 v = hp[lane];
    float* op = cur + (size_t)d * FSTRIDE + lane * 4;
    atomicAdd(op + 0, v.x * ns);
    atomicAdd(op + 1, v.y * ns);
    atomicAdd(op + 2, v.z * ns);
    atomicAdd(op + 3, v.w * ns);
}

// ---------------- WMMA f32 GEMM: Out(MxN) = A(MxK) @ W(KxN) + bias ----------
// A row-major with lda == K. One block per 16-row tile; NTILES waves, each
// owning one 16x16 output tile. A tile staged in LDS (padded, conflict-free).
// V_WMMA_F32_16X16X4_F32 VGPR layouts per CDNA5 ISA 7.12.2:
//   A 16x4:  lanes 0-15 hold K=k,k+1 ; lanes 16-31 hold K=k+2,k+3
//   B 4x16:  mirrored (rows striped across lanes within a VGPR)
//   C 16x16: VGPR r: lanes 0-15 -> (M=r, N=lane); lanes 16-31 -> (M=8+r)

template <int K, int NTILES>
__global__ void tag_gemm_wmma(const float* __restrict__ A,
                              const float* __restrict__ W,
                              const float* __restrict__ bias,
                              float* __restrict__ Out,
                              int M) {
    constexpr int N  = NTILES * 16;
    constexpr int KP = K + 4;                 // pad: LDS bank-conflict free
    __shared__ float As[16 * KP];

    const int tid  = threadIdx.x;
    const int wave = tid >> 5;                // 0..NTILES-1 -> n-tile
    const int lane = tid & 31;
    const int half = lane >> 4;               // 0 or 1
    const int l    = lane & 15;
    const int m0   = blockIdx.x * 16;

    // cooperative coalesced load of the 16xK A tile
    constexpr int NV = 16 * K / 4;            // float4 elements
    for (int idx = tid; idx < NV; idx += NTILES * 32) {
        int r = idx / (K / 4);
        int c = (idx % (K / 4)) * 4;
        float4 v = *(const float4*)(A + (size_t)(m0 + r) * K + c);
        float* s = &As[r * KP + c];
        s[0] = v.x; s[1] = v.y; s[2] = v.z; s[3] = v.w;
    }
    __syncthreads();

    const int n0 = wave * 16;
    v8f acc = {};
#pragma unroll 4
    for (int k = 0; k < K; k += 4) {
        int kk = k + half * 2;
        v2f a, b;
        a.x = As[l * KP + kk];
        a.y = As[l * KP + kk + 1];
        b.x = W[(size_t)kk * N + n0 + l];
        b.y = W[(size_t)(kk + 1) * N + n0 + l];
        acc = __builtin_amdgcn_wmma_f32_16x16x4_f32(
            /*neg_a=*/false, a, /*neg_b=*/false, b,
            /*c_mod=*/(short)0, acc, /*reuse_a=*/false, /*reuse_b=*/false);
    }

    const float bv = bias[n0 + l];
#pragma unroll
    for (int r = 0; r < 8; ++r) {
        int row = m0 + r + half * 8;
        Out[(size_t)row * N + n0 + l] = acc[r] + bv;
    }
}

// ---------------- host-side launch ----------------

extern "C" void kernel_launch(void* const* d_in, const int* in_sizes, int n_in,
                              void* d_out, int out_size, void* d_ws, size_t ws_size,
                              hipStream_t stream) {
    const float* x   = (const float*)d_in[0];
    const int*   src = (const int*)  d_in[1];
    const int*   dst = (const int*)  d_in[2];
    const float* W1  = (const float*)d_in[3];
    const float* b1  = (const float*)d_in[4];
    const float* W2  = (const float*)d_in[5];
    const float* b2  = (const float*)d_in[6];
    const float* Wfc = (const float*)d_in[7];
    const float* bfc = (const float*)d_in[8];
    float* out = (float*)d_out;

    // workspace layout
    char* ws = (char*)d_ws;
    float* normbuf = (float*)ws;                                   // 50000 f32
    size_t off = ((size_t)TAG_NODES * 4 + 255) & ~(size_t)255;
    float* fstack = (float*)(ws + off);                            // 50000 x 512
    float* hbuf   = fstack + (size_t)TAG_NODES * FSTRIDE;          // 50000 x 128

    const int nElems  = TAG_NODES * TAG_D;                         // 6.4M
    const int ebBlks  = (nElems + 255) / 256;                      // 25000
    const int nodeBlk = (TAG_NODES + 255) / 256;
    const int edgeBlk = (TAG_EDGES + 255) / 256;
    const int scatBlk = (TAG_EDGES * 32 + 255) / 256;              // wave/edge

    // degree -> norm
    tag_zero_f32<<<nodeBlk, 256, 0, stream>>>(normbuf, TAG_NODES);
    tag_deg<<<edgeBlk, 256, 0, stream>>>(dst, normbuf, TAG_EDGES);
    tag_norm_finalize<<<nodeBlk, 256, 0, stream>>>(normbuf, TAG_NODES);

    auto run_hops = [&](const float* inp) {
        tag_copy_to_slot<<<ebBlks, 256, 0, stream>>>(inp, fstack, nElems);
        for (int h = 1; h <= 3; ++h) {
            float* cur = fstack + h * TAG_D;
            const float* prev = fstack + (h - 1) * TAG_D;
            tag_zero_slot<<<ebBlks, 256, 0, stream>>>(cur, nElems);
            tag_edge_scatter<<<scatBlk, 256, 0, stream>>>(src, dst, normbuf, prev, cur, TAG_EDGES);
            tag_scale_slot<<<ebBlks, 256, 0, stream>>>(cur, normbuf, nElems);
        }
    };

    const int mTiles = TAG_NODES / 16;  // 3125 (exact)

    // layer 1
    run_hops(x);
    tag_gemm_wmma<512, 8><<<mTiles, 8 * 32, 0, stream>>>(fstack, W1, b1, hbuf, TAG_NODES);
    // layer 2
    run_hops(hbuf);
    tag_gemm_wmma<512, 8><<<mTiles, 8 * 32, 0, stream>>>(fstack, W2, b2, hbuf, TAG_NODES);
    // fc
    tag_gemm_wmma<128, 4><<<mTiles, 4 * 32, 0, stream>>>(hbuf, Wfc, bfc, out, TAG_NODES);
}